// GCN_AutoEncoder_90314572300354
// MI455X (gfx1250) — compile-verified
//
#include <hip/hip_runtime.h>
#include <cmath>

#define N_NODES 20000
#define N_EDGES 320000
#define F_IN    256
#define N_HID   128
#define MIDW    800

typedef __attribute__((ext_vector_type(2))) float v2f;
typedef __attribute__((ext_vector_type(8))) float v8f;

__device__ __forceinline__ float sigmoidf_(float v) {
    return 1.0f / (1.0f + __expf(-v));
}

// ---------------- degree + norm ----------------
__global__ void deg_kernel(const int* __restrict__ src, const int* __restrict__ dst,
                           float* __restrict__ degs, float* __restrict__ degd) {
    int i = blockIdx.x * blockDim.x + threadIdx.x;
    if (i < N_EDGES) {
        unsafeAtomicAdd(&degs[src[i]], 1.0f);
        unsafeAtomicAdd(&degd[dst[i]], 1.0f);
    }
}

__global__ void norm_kernel(float* __restrict__ ns, float* __restrict__ nd) {
    int i = blockIdx.x * blockDim.x + threadIdx.x;
    if (i < N_NODES) {
        ns[i] = rsqrtf(fmaxf(ns[i], 1.0f));
        nd[i] = rsqrtf(fmaxf(nd[i], 1.0f));
    }
}

// ---------------- row scaling: out[i,:] = x[i,:] * s[i] ----------------
template <int W4>  // width in float4 units (power of two)
__global__ void scale_rows_kernel(const float* __restrict__ x, const float* __restrict__ s,
                                  float* __restrict__ out) {
    int i = blockIdx.x * blockDim.x + threadIdx.x;
    if (i >= N_NODES * W4) return;
    int row = i / W4;
    float4 v = ((const float4*)x)[i];
    float sc = s[row];
    v.x *= sc; v.y *= sc; v.z *= sc; v.w *= sc;
    ((float4*)out)[i] = v;
}

// ---------------- edge scatter-add: agg[dst[e],:] += t[src[e],:] ----------------
template <int W4>  // width in float4 units
__global__ void scatter_kernel(const float* __restrict__ t, const int* __restrict__ srcI,
                               const int* __restrict__ dstI, float* __restrict__ agg) {
    int i = blockIdx.x * blockDim.x + threadIdx.x;
    if (i >= N_EDGES * W4) return;
    int e = i / W4;
    int c = i % W4;
    int s = srcI[e];
    int d = dstI[e];
    float4 v = ((const float4*)t)[(size_t)s * W4 + c];
    float* p = agg + ((size_t)d * W4 + c) * 4;
    unsafeAtomicAdd(p + 0, v.x);
    unsafeAtomicAdd(p + 1, v.y);
    unsafeAtomicAdd(p + 2, v.z);
    unsafeAtomicAdd(p + 3, v.w);
}

// ---------------- fp32 WMMA GEMM: C = epilogue(A[MxK] @ B[KxN]) ----------------
// EPI==0: C = acc * scale[row]                      (pre-aggregation ns scaling)
// EPI==1: C = relu(acc * scale[row] + bias[col])    (post-aggregation nd + bias + relu)
// Wave tile: 32 rows (2 x 16, MT=2) x NT*16 cols. Block: 8 waves -> 256 rows.
// Requirements: M % 32 == 0, K % 4 == 0, columns [ncol0, ncol0 + gridDim.x*NT*16) valid.
template <int EPI, int NT>
__global__ __launch_bounds__(256) void gemm_wmma_kernel(
    const float* __restrict__ A, const float* __restrict__ B,
    const float* __restrict__ scale, const float* __restrict__ bias,
    float* __restrict__ C, int M, int N, int K, int ncol0) {
    const int lane = threadIdx.x & 31;
    const int wave = threadIdx.x >> 5;
    const int m0 = (blockIdx.y * 8 + wave) * 32;
    const int n0 = ncol0 + blockIdx.x * (NT * 16);
    if (m0 + 32 > M) return;  // wave-uniform: EXEC stays all-ones for WMMA

    // Fragment lane mapping for 32-bit 16x4 A / 4x16 B (ISA 7.12.2):
    // lanes 0-15 carry K = {k, k+1}; lanes 16-31 carry K = {k+2, k+3}.
    const int mr = lane & 15;           // M index (A) / N sub-index (B)
    const int koff = (lane >> 4) << 1;  // 0 or 2

    v8f acc[2][NT] = {};

    const float* Ap0 = A + (size_t)(m0 + mr) * K + koff;        // rows m0..m0+15
    const float* Ap1 = Ap0 + (size_t)16 * K;                    // rows m0+16..m0+31
    const float* Bp = B + (size_t)koff * N + n0 + mr;

#pragma unroll 2
    for (int k = 0; k < K; k += 4) {
        // Issue ALL fragment loads for this k-step first (distinct registers),
        // so one wait covers the whole WMMA burst and next-step loads overlap.
        v2f a0, a1, bb[NT];
        a0.x = Ap0[k];      a0.y = Ap0[k + 1];
        a1.x = Ap1[k];      a1.y = Ap1[k + 1];
        const float* Bk = Bp + (size_t)k * N;
#pragma unroll
        for (int t = 0; t < NT; ++t) {
            bb[t].x = Bk[t * 16];      // B[k+koff][n]
            bb[t].y = Bk[t * 16 + N];  // B[k+koff+1][n]
        }
#pragma unroll
        for (int t = 0; t < NT; ++t) {
            acc[0][t] = __builtin_amdgcn_wmma_f32_16x16x4_f32(
                false, a0, false, bb[t], (short)0, acc[0][t], false, false);
            acc[1][t] = __builtin_amdgcn_wmma_f32_16x16x4_f32(
                false, a1, false, bb[t], (short)0, acc[1][t], false, false);
        }
    }

    // C fragment: VGPR r -> row base + r + (lane>=16 ? 8 : 0), col n0 + 16*t + (lane&15)
    const int rbase = (lane >> 4) << 3;
#pragma unroll
    for (int mi = 0; mi < 2; ++mi) {
#pragma unroll
        for (int t = 0; t < NT; ++t) {
            int c = n0 + t * 16 + mr;
            float bv = (EPI == 1) ? bias[c] : 0.0f;
#pragma unroll
            for (int r = 0; r < 8; ++r) {
                int row = m0 + mi * 16 + rbase + r;
                float v = acc[mi][t][r] * scale[row];
                if (EPI == 1) v = fmaxf(v + bv, 0.0f);
                C[(size_t)row * N + c] = v;
            }
        }
    }
}

// Host-side splitter: full 64-wide blocks with NT=4, 32-wide remainder with NT=2.
template <int EPI>
static void gemm_launch(const float* A, const float* B, const float* scl, const float* bias,
                        float* C, int M, int N, int K, hipStream_t stream) {
    int gy = (M + 255) / 256;
    int nb64 = N / 64;
    if (nb64 > 0) {
        dim3 g(nb64, gy);
        gemm_wmma_kernel<EPI, 4><<<g, 256, 0, stream>>>(A, B, scl, bias, C, M, N, K, 0);
    }
    if (N - nb64 * 64 >= 32) {  // all our N are multiples of 32
        dim3 g(1, gy);
        gemm_wmma_kernel<EPI, 2><<<g, 256, 0, stream>>>(A, B, scl, bias, C, M, N, K, nb64 * 64);
    }
}

// ---------------- finalize kernels ----------------
__global__ void fin_enc_kernel(const float* __restrict__ agg, const float* __restrict__ nd,
                               const float* __restrict__ b, float* __restrict__ h2,
                               float* __restrict__ enc) {
    int i = blockIdx.x * blockDim.x + threadIdx.x;
    if (i >= N_NODES * N_HID) return;
    int row = i >> 7;  // /128
    int col = i & 127;
    float v = agg[i] * nd[row] + b[col];
    h2[i] = v;              // pre-sigmoid feeds layer 3
    enc[i] = sigmoidf_(v);  // encoded output
}

__global__ void fin_dec_kernel(const float* __restrict__ agg, const float* __restrict__ nd,
                               const float* __restrict__ b, float* __restrict__ dec) {
    int i = blockIdx.x * blockDim.x + threadIdx.x;
    if (i >= N_NODES * F_IN) return;
    int row = i >> 8;  // /256
    int col = i & 255;
    dec[i] = sigmoidf_(agg[i] * nd[row] + b[col]);
}

// ---------------- launch ----------------
extern "C" void kernel_launch(void* const* d_in, const int* in_sizes, int n_in,
                              void* d_out, int out_size, void* d_ws, size_t ws_size,
                              hipStream_t stream) {
    const float* x  = (const float*)d_in[0];
    const int* src  = (const int*)d_in[1];
    const int* dst  = (const int*)d_in[2];
    const float* W1 = (const float*)d_in[3];
    const float* b1 = (const float*)d_in[4];
    const float* W2 = (const float*)d_in[5];
    const float* b2 = (const float*)d_in[6];
    const float* W3 = (const float*)d_in[7];
    const float* b3 = (const float*)d_in[8];
    const float* W4 = (const float*)d_in[9];
    const float* b4 = (const float*)d_in[10];

    float* enc = (float*)d_out;                  // [20000, 128]
    float* dec = enc + (size_t)N_NODES * N_HID;  // [20000, 256]

    float* ws   = (float*)d_ws;
    float* ns   = ws;                            // 20000
    float* nd   = ns + N_NODES;                  // 20000
    float* t    = nd + N_NODES;                  // 20000*256 (max pre-agg width)
    float* agg  = t + (size_t)N_NODES * F_IN;    // 20000*256
    float* hmid = agg + (size_t)N_NODES * F_IN;  // 20000*800
    float* h2   = hmid + (size_t)N_NODES * MIDW; // 20000*128

    auto cdiv = [](int a, int b) { return (a + b - 1) / b; };
    const int BT = 256;

    // degrees -> rsqrt norms
    hipMemsetAsync(ns, 0, sizeof(float) * 2 * N_NODES, stream);
    deg_kernel<<<cdiv(N_EDGES, BT), BT, 0, stream>>>(src, dst, ns, nd);
    norm_kernel<<<cdiv(N_NODES, BT), BT, 0, stream>>>(ns, nd);

    // ---- Layer 1: t = x*ns ; agg = scatter(t) ; hmid = relu((agg@W1)*nd + b1)
    scale_rows_kernel<64><<<cdiv(N_NODES * 64, BT), BT, 0, stream>>>(x, ns, t);
    hipMemsetAsync(agg, 0, sizeof(float) * (size_t)N_NODES * F_IN, stream);
    scatter_kernel<64><<<cdiv(N_EDGES * 64, BT), BT, 0, stream>>>(t, src, dst, agg);
    gemm_launch<1>(agg, W1, nd, b1, hmid, N_NODES, MIDW, F_IN, stream);

    // ---- Layer 2: t = (hmid@W2)*ns ; agg = scatter(t) ; h2 = agg*nd + b2 ; enc = sigmoid(h2)
    gemm_launch<0>(hmid, W2, ns, nullptr, t, N_NODES, N_HID, MIDW, stream);
    hipMemsetAsync(agg, 0, sizeof(float) * (size_t)N_NODES * N_HID, stream);
    scatter_kernel<32><<<cdiv(N_EDGES * 32, BT), BT, 0, stream>>>(t, src, dst, agg);
    fin_enc_kernel<<<cdiv(N_NODES * N_HID, BT), BT, 0, stream>>>(agg, nd, b2, h2, enc);

    // ---- Layer 3: t = h2*ns ; agg = scatter(t) ; hmid = relu((agg@W3)*nd + b3)
    scale_rows_kernel<32><<<cdiv(N_NODES * 32, BT), BT, 0, stream>>>(h2, ns, t);
    hipMemsetAsync(agg, 0, sizeof(float) * (size_t)N_NODES * N_HID, stream);
    scatter_kernel<32><<<cdiv(N_EDGES * 32, BT), BT, 0, stream>>>(t, src, dst, agg);
    gemm_launch<1>(agg, W3, nd, b3, hmid, N_NODES, MIDW, N_HID, stream);

    // ---- Layer 4: t = (hmid@W4)*ns ; agg = scatter(t) ; dec = sigmoid(agg*nd + b4)
    gemm_launch<0>(hmid, W4, ns, nullptr, t, N_NODES, F_IN, MIDW, stream);
    hipMemsetAsync(agg, 0, sizeof(float) * (size_t)N_NODES * F_IN, stream);
    scatter_kernel<64><<<cdiv(N_EDGES * 64, BT), BT, 0, stream>>>(t, src, dst, agg);
    fin_dec_kernel<<<cdiv(N_NODES * F_IN, BT), BT, 0, stream>>>(agg, nd, b4, dec);
}